// EpisodicMemory_22840636080828
// MI455X (gfx1250) — compile-verified
//
#include <hip/hip_runtime.h>
#include <math.h>

typedef float v2f __attribute__((ext_vector_type(2)));
typedef float v8f __attribute__((ext_vector_type(8)));

#define NROW 2048
#define DDIM 512
#define NBATCH 8
#define LDA 516   // padded LDS row stride (floats): 516 % 64 = 4 -> conflict-free

// ---------------------------------------------------------------------------
// Kernel 0: per-batch cumsum of boundaries -> community ids
// ---------------------------------------------------------------------------
__global__ void comm_kernel(const int* __restrict__ bnd, int* __restrict__ comm) {
    const int b = blockIdx.x;
    if (threadIdx.x == 0) {
        const int* src = bnd + b * NROW;
        int* dst = comm + b * NROW;
        int s = 0;
        for (int i = 0; i < NROW; ++i) { s += src[i]; dst[i] = s; }
    }
}

// ---------------------------------------------------------------------------
// Kernel 1: fused Gram pass with 2x2 register blocking.
// Block = 128 threads (4 waves) handles a 32-row panel of one batch.
// Each wave computes a 32x32 macro-tile per step: 2 A loads + 2 B loads feed
// 4 x V_WMMA_F32_16X16X4_F32 (halves memory ops per matrix op vs 1x1).
// Emits rowsum_i = sum_j S_ij and sqsum_i = sum_j S_ij^2, S never stored.
// ---------------------------------------------------------------------------
__global__ __launch_bounds__(128) void gram_stats_kernel(const float* __restrict__ rep,
                                                         float* __restrict__ rowsum,
                                                         float* __restrict__ sqsum) {
    const int mtile = blockIdx.x;   // 0..63 (32-row super-tile)
    const int batch = blockIdx.y;   // 0..7
    const float* R = rep + (size_t)batch * NROW * DDIM;
    const int m0 = mtile * 32;

    __shared__ float As[32 * LDA];
    __shared__ float redR[4][32];
    __shared__ float redQ[4][32];

    const int t = threadIdx.x;

    // cooperatively stage the 32 x 512 fp32 A panel into LDS (float4 loads)
    for (int chunk = 0; chunk < 32; ++chunk) {
        const int f   = t + 128 * chunk;   // float4 index 0..4095
        const int row = f >> 7;
        const int col = (f & 127) << 2;
        const float4 v = *(const float4*)(R + (size_t)(m0 + row) * DDIM + col);
        float* dst = &As[row * LDA + col];
        dst[0] = v.x; dst[1] = v.y; dst[2] = v.z; dst[3] = v.w;
    }
    __syncthreads();

    const int wave = t >> 5;        // 0..3
    const int lane = t & 31;
    const int lrow = lane & 15;     // row (A) / col (B) owned by this lane
    const int hi   = lane >> 4;     // K sub-chunk select per A/B 16x4 layout

    float rs[2][8], qs[2][8];
#pragma unroll
    for (int m = 0; m < 2; ++m)
#pragma unroll
        for (int r = 0; r < 8; ++r) { rs[m][r] = 0.f; qs[m][r] = 0.f; }

    const float* a0p = &As[lrow * LDA + 2 * hi];
    const float* a1p = &As[(lrow + 16) * LDA + 2 * hi];

    // waves own interleaved pairs of n-tiles: {2w,2w+1}, {2w+8,2w+9}, ...
    for (int base = wave * 2; base < NROW / 16; base += 8) {
        const int n0 = base * 16;
        const int n1 = n0 + 16;
        v8f c00 = {}, c01 = {}, c10 = {}, c11 = {};
        const float* b0p = R + (size_t)(n0 + lrow) * DDIM + 2 * hi;
        const float* b1p = R + (size_t)(n1 + lrow) * DDIM + 2 * hi;
#pragma unroll 4
        for (int kk = 0; kk < DDIM; kk += 4) {
            v2f a0 = *(const v2f*)(a0p + kk);
            v2f a1 = *(const v2f*)(a1p + kk);
            v2f b0 = *(const v2f*)(b0p + kk);
            v2f b1 = *(const v2f*)(b1p + kk);
            c00 = __builtin_amdgcn_wmma_f32_16x16x4_f32(false, a0, false, b0, (short)0, c00, false, false);
            c01 = __builtin_amdgcn_wmma_f32_16x16x4_f32(false, a0, false, b1, (short)0, c01, false, false);
            c10 = __builtin_amdgcn_wmma_f32_16x16x4_f32(false, a1, false, b0, (short)0, c10, false, false);
            c11 = __builtin_amdgcn_wmma_f32_16x16x4_f32(false, a1, false, b1, (short)0, c11, false, false);
        }
#pragma unroll
        for (int r = 0; r < 8; ++r) {
            rs[0][r] += c00[r] + c01[r];
            qs[0][r] += c00[r] * c00[r] + c01[r] * c01[r];
            rs[1][r] += c10[r] + c11[r];
            qs[1][r] += c10[r] * c10[r] + c11[r] * c11[r];
        }
    }

    // butterfly reduce across the 16 lanes of each half (C tile: lanes 0-15
    // cols N, VGPR r -> row r; lanes 16-31 cols N, VGPR r -> row 8+r)
#pragma unroll
    for (int m = 0; m < 2; ++m) {
#pragma unroll
        for (int r = 0; r < 8; ++r) {
            float a = rs[m][r], q = qs[m][r];
            for (int off = 1; off < 16; off <<= 1) {
                a += __shfl_xor(a, off, 32);
                q += __shfl_xor(q, off, 32);
            }
            rs[m][r] = a; qs[m][r] = q;
        }
    }
    if (lrow == 0) {
#pragma unroll
        for (int m = 0; m < 2; ++m)
#pragma unroll
            for (int r = 0; r < 8; ++r) {
                redR[wave][m * 16 + hi * 8 + r] = rs[m][r];
                redQ[wave][m * 16 + hi * 8 + r] = qs[m][r];
            }
    }
    __syncthreads();

    if (t < 32) {
        const float a = redR[0][t] + redR[1][t] + redR[2][t] + redR[3][t];
        const float q = redQ[0][t] + redQ[1][t] + redQ[2][t] + redQ[3][t];
        rowsum[(size_t)batch * NROW + m0 + t] = a;
        sqsum [(size_t)batch * NROW + m0 + t] = q;
    }
}

// ---------------------------------------------------------------------------
// Kernel 2: per-row within-segment contribution.
// comm is non-decreasing -> same-community pairs are contiguous; recompute only
// the tiny block-diagonal S entries (avg segment length ~2).
// rowcontrib_i = (sum_{j in seg(i)} rep_i . rep_j) / max(||S_i||, 1e-12)
// One wave per row; 8 waves per block.
// ---------------------------------------------------------------------------
__global__ __launch_bounds__(256) void within_kernel(const float* __restrict__ rep,
                                                     const int* __restrict__ comm,
                                                     const float* __restrict__ sqsum,
                                                     float* __restrict__ rowcontrib) {
    const int wave = threadIdx.x >> 5;
    const int lane = threadIdx.x & 31;
    const int g = blockIdx.x * 8 + wave;        // 0 .. 8*2048-1
    const int batch = g >> 11;
    const int i = g & (NROW - 1);

    const float* R  = rep  + (size_t)batch * NROW * DDIM;
    const int*   cm = comm + batch * NROW;
    const int ci = cm[i];

    int s = i; while (s > 0        && cm[s - 1] == ci) --s;
    int e = i; while (e < NROW - 1 && cm[e + 1] == ci) ++e;

    const float* a = R + (size_t)i * DDIM;
    float acc = 0.f;
    for (int j = s; j <= e; ++j) {
        const float* b = R + (size_t)j * DDIM;
        float d = 0.f;
        for (int k = lane; k < DDIM; k += 32) d += a[k] * b[k];
        acc += d;
    }
    for (int off = 1; off < 32; off <<= 1) acc += __shfl_xor(acc, off, 32);

    if (lane == 0) {
        const float nrm = fmaxf(sqrtf(sqsum[(size_t)batch * NROW + i]), 1e-12f);
        rowcontrib[(size_t)batch * NROW + i] = acc / nrm;
    }
}

// ---------------------------------------------------------------------------
// Kernel 3: per-batch combine (serial; O(N), negligible). Deterministic order.
// ---------------------------------------------------------------------------
__global__ void final_kernel(const int* __restrict__ comm,
                             const float* __restrict__ rowsum,
                             const float* __restrict__ sqsum,
                             const float* __restrict__ rowcontrib,
                             float* __restrict__ out) {
    const int batch = blockIdx.x;
    __shared__ float s_within[NROW + 1];
    __shared__ float s_degsum[NROW + 1];
    __shared__ int   s_cnt[NROW + 1];

    if (threadIdx.x != 0) return;

    const int*   cm   = comm       + batch * NROW;
    const float* rsum = rowsum     + (size_t)batch * NROW;
    const float* qsum = sqsum      + (size_t)batch * NROW;
    const float* rc   = rowcontrib + (size_t)batch * NROW;

    float total = 0.f;
    for (int i = 0; i < NROW; ++i) {
        const float nrm = fmaxf(sqrtf(qsum[i]), 1e-12f);
        total += rsum[i] / nrm;                       // total = sum_ij S_hat
    }
    for (int c = 0; c <= NROW; ++c) { s_within[c] = 0.f; s_degsum[c] = 0.f; s_cnt[c] = 0; }

    float sum_within = 0.f;
    for (int i = 0; i < NROW; ++i) {
        const float nrm = fmaxf(sqrtf(qsum[i]), 1e-12f);
        const float deg = rsum[i] / nrm;
        const int c = cm[i];
        s_within[c] += rc[i];
        s_degsum[c] += deg;
        s_cnt[c]    += 1;
        sum_within  += rc[i];
    }

    float sumdeg2 = 0.f, condacc = 0.f;
    int ncomm = 0;
    for (int c = 0; c <= NROW; ++c) {
        sumdeg2 += s_degsum[c] * s_degsum[c];         // sum_same deg_i*deg_j
        if (s_cnt[c] > 0) {
            const float between = s_degsum[c] - s_within[c];
            condacc += between / (2.f * s_within[c] + between + 1e-10f);
            ++ncomm;
        }
    }
    const float mod  = (sum_within - sumdeg2 / total) / total;
    const float cond = condacc / (float)(ncomm > 0 ? ncomm : 1);
    out[batch]          = mod;   // row 0: modularity
    out[NBATCH + batch] = cond;  // row 1: conductance
}

// ---------------------------------------------------------------------------
extern "C" void kernel_launch(void* const* d_in, const int* in_sizes, int n_in,
                              void* d_out, int out_size, void* d_ws, size_t ws_size,
                              hipStream_t stream) {
    const float* rep = (const float*)d_in[0];   // (8, 2048, 512) f32
    const int*   bnd = (const int*)d_in[1];     // (8, 2048) i32
    float* out = (float*)d_out;                 // (2, 8) f32

    char* ws = (char*)d_ws;
    int*   comm       = (int*)  (ws + 0);        // 8*2048*4 = 64KB
    float* rowsum     = (float*)(ws + 65536);    // 64KB
    float* sqsum      = (float*)(ws + 131072);   // 64KB
    float* rowcontrib = (float*)(ws + 196608);   // 64KB  (total 256KB)

    comm_kernel<<<NBATCH, 32, 0, stream>>>(bnd, comm);

    dim3 g1(NROW / 32, NBATCH);                  // 64 row-panels x 8 batches
    gram_stats_kernel<<<g1, 128, 0, stream>>>(rep, rowsum, sqsum);

    within_kernel<<<(NBATCH * NROW) / 8, 256, 0, stream>>>(rep, comm, sqsum, rowcontrib);

    final_kernel<<<NBATCH, 32, 0, stream>>>(comm, rowsum, sqsum, rowcontrib, out);
}